// BBPMMemoryFloat_26474178413369
// MI455X (gfx1250) — compile-verified
//
#include <hip/hip_runtime.h>
#include <hip/hip_bf16.h>
#include <math.h>

// ---------------------------------------------------------------------------
// BBPM hashed memory: scatter-add (write) + debiased mean-pool gather (read).
//
// MI455X reasoning: 128MB table + 1MB counts are L2-resident (192MB L2), so
// the op is bound by L2 fp32-atomic throughput and L2 gather bandwidth — WMMA
// is inapplicable (no shared matmul operand). Chosen gfx1250 paths:
//   * global_atomic_add_f32 (unsafeAtomicAdd) for the 134M scatter adds
//   * s_load (SMEM) for wave-uniform slot indices / counts (no ds_bpermute)
//   * 512B-coalesced global_load_b128 gathers (wave32: lane = float4 column)
//   * global_prefetch_b8 to stream upcoming gathered rows into the WGP cache
// ---------------------------------------------------------------------------

// ---- Kernel 1: initialize workspace table from (immutable) inputs ----------
__global__ void bbpm_init_ws(const float4* __restrict__ mem_in,
                             const float*  __restrict__ cnt_in,
                             float4* __restrict__ mem_ws,
                             float*  __restrict__ cnt_ws,
                             size_t n_mem4, size_t n_cnt) {
    size_t i      = (size_t)blockIdx.x * blockDim.x + threadIdx.x;
    size_t stride = (size_t)gridDim.x * blockDim.x;
    for (size_t k = i; k < n_mem4; k += stride) mem_ws[k] = mem_in[k];   // b128 copies
    for (size_t k = i; k < n_cnt;  k += stride) cnt_ws[k] = cnt_in[k];
}

// ---- Kernel 2 (specialized): scatter-add ----------------------------------
// One block (= DDIM threads) per key b. row[s] is wave-uniform -> the fully
// unrolled index reads become batched scalar s_loads; each thread then issues
// one global_atomic_add_f32 per slot into the L2-resident table.
template<int KH, int DDIM>
__global__ void bbpm_scatter_t(const int*   __restrict__ idx,
                               const float* __restrict__ vals,
                               float* __restrict__ mem_ws,
                               float* __restrict__ cnt_ws,
                               float scale) {
    const int b = blockIdx.x;
    const int t = threadIdx.x;                      // column 0..DDIM-1
    const int* __restrict__ row = idx + (size_t)b * KH;

    const float v = vals[(size_t)b * DDIM + t] * scale;

    #pragma unroll
    for (int s = 0; s < KH; ++s) {
        const int j = row[s];                       // uniform -> s_load
        unsafeAtomicAdd(&mem_ws[(size_t)j * DDIM + t], v);  // global_atomic_add_f32
    }
    if (t < KH) {                                   // one count bump per slot
        unsafeAtomicAdd(&cnt_ws[row[t]], 1.0f);
    }
}

// ---- Kernel 3 (specialized): gather + debias + mean-pool ------------------
// One wave per key b (blockDim = {32, 8}). Lane l owns float4 column chunk l
// (32 chunks * 4 = DDIM = 128). Indices and counts are wave-uniform scalar
// loads; row data is one 512B global_load_b128 per wave per slot, with
// global_prefetch_b8 running 4 slots ahead.
template<int KH, int DDIM>
__global__ void bbpm_gather_t(const int*   __restrict__ idx,
                              const float* __restrict__ mem_ws,
                              const float* __restrict__ cnt_ws,
                              float* __restrict__ out,
                              int B) {
    const int b = blockIdx.x * blockDim.y + threadIdx.y;
    if (b >= B) return;
    const int lane = threadIdx.x;                   // 0..31, float4 chunk owner
    const int* __restrict__ row = idx + (size_t)b * KH;

    float4 acc = make_float4(0.f, 0.f, 0.f, 0.f);

    #pragma unroll 8
    for (int s = 0; s < KH; ++s) {
        if (s + 4 < KH) {                           // stream 4 slots ahead
            __builtin_prefetch(mem_ws + (size_t)row[s + 4] * DDIM + lane * 4, 0, 1);
        }
        const int   j = row[s];                     // uniform -> s_load
        const float w = 1.0f / (cnt_ws[j] + 1e-8f); // uniform debias weight
        const float4 m =
            ((const float4*)(mem_ws + (size_t)j * DDIM))[lane];  // global_load_b128
        acc.x += m.x * w;
        acc.y += m.y * w;
        acc.z += m.z * w;
        acc.w += m.w * w;
    }

    constexpr float inv_kh = 1.0f / (float)KH;
    float4 o = make_float4(acc.x * inv_kh, acc.y * inv_kh,
                           acc.z * inv_kh, acc.w * inv_kh);
    ((float4*)(out + (size_t)b * DDIM))[lane] = o;  // global_store_b128
}

// ---- Generic fallbacks (any KH, d) ----------------------------------------
__global__ void bbpm_scatter_g(const int*   __restrict__ idx,
                               const float* __restrict__ vals,
                               float* __restrict__ mem_ws,
                               float* __restrict__ cnt_ws,
                               int KH, int d, float scale) {
    const int b = blockIdx.x;
    const int t = threadIdx.x;
    const int* __restrict__ row = idx + (size_t)b * KH;
    const float v = vals[(size_t)b * d + t] * scale;
    for (int s = 0; s < KH; ++s)
        unsafeAtomicAdd(&mem_ws[(size_t)row[s] * d + t], v);
    if (t < KH)
        unsafeAtomicAdd(&cnt_ws[row[t]], 1.0f);
}

__global__ void bbpm_gather_g(const int*   __restrict__ idx,
                              const float* __restrict__ mem_ws,
                              const float* __restrict__ cnt_ws,
                              float* __restrict__ out,
                              int KH, int d, float inv_kh) {
    const int b = blockIdx.x;
    const int t = threadIdx.x;
    const int* __restrict__ row = idx + (size_t)b * KH;
    float acc = 0.0f;
    for (int s = 0; s < KH; ++s) {
        const int   j = row[s];
        const float w = 1.0f / (cnt_ws[j] + 1e-8f);
        acc += mem_ws[(size_t)j * d + t] * w;
    }
    out[(size_t)b * d + t] = acc * inv_kh;
}

// ---------------------------------------------------------------------------
extern "C" void kernel_launch(void* const* d_in, const int* in_sizes, int n_in,
                              void* d_out, int out_size, void* d_ws, size_t ws_size,
                              hipStream_t stream) {
    const int*   indices = (const int*)  d_in[0];   // [B, KH] int32
    const float* values  = (const float*)d_in[1];   // [B, d]  f32
    const float* memory  = (const float*)d_in[2];   // [D, d]  f32 (immutable)
    const float* counts  = (const float*)d_in[3];   // [D, 1]  f32 (immutable)

    // Derive shapes (reference: D=262144, d=128, B=32768, KH=32).
    const int D  = in_sizes[3];
    const int d  = in_sizes[2] / D;
    const int B  = in_sizes[1] / d;
    const int KH = in_sizes[0] / B;

    const float scale  = 1.0f / sqrtf((float)KH);
    const float inv_kh = 1.0f / (float)KH;

    // Workspace: fp32 accumulation table + counts (inputs stay untouched).
    float* mem_ws = (float*)d_ws;                   // D*d floats (128 MB)
    float* cnt_ws = mem_ws + (size_t)D * d;         // D floats   (1 MB)

    const size_t n_mem4 = ((size_t)D * d) / 4;
    bbpm_init_ws<<<2048, 256, 0, stream>>>((const float4*)memory, counts,
                                           (float4*)mem_ws, cnt_ws,
                                           n_mem4, (size_t)D);

    if (KH == 32 && d == 128) {
        bbpm_scatter_t<32, 128><<<B, 128, 0, stream>>>(indices, values,
                                                       mem_ws, cnt_ws, scale);
        dim3 gblk(32, 8);                           // 8 keys per 256-thread block
        int  gblocks = (B + 7) / 8;
        bbpm_gather_t<32, 128><<<gblocks, gblk, 0, stream>>>(indices, mem_ws,
                                                             cnt_ws, (float*)d_out, B);
    } else {
        bbpm_scatter_g<<<B, d, 0, stream>>>(indices, values, mem_ws, cnt_ws,
                                            KH, d, scale);
        bbpm_gather_g<<<B, d, 0, stream>>>(indices, mem_ws, cnt_ws,
                                           (float*)d_out, KH, d, inv_kh);
    }
}